// GeneralHMUStack_49520972923461
// MI455X (gfx1250) — compile-verified
//
#include <hip/hip_runtime.h>
#include <hip/hip_bf16.h>

// ---------------------------------------------------------------------------
// GeneralHMUStack for MI455X (gfx1250, wave32, WMMA f32 16x16x4)
// B = D = N = 1024, K = 8, L = 3, C = 10
// One wave computes a 32(b) x 16(n) output block: 18 f32 WMMA accumulators
// (2x S tiles + 2x8 P tiles), so each 9-fragment weight load feeds 18 WMMAs.
// ---------------------------------------------------------------------------

#define DIM   1024
#define NB    1024
#define KK    8
#define NCLS  10
#define BN_EPS 1e-5f

typedef __attribute__((ext_vector_type(2))) float v2f;
typedef __attribute__((ext_vector_type(8))) float v8f;

__device__ __forceinline__ v8f wmma_f32_4(v2f a, v2f b, v8f c) {
  // D = A(16x4,f32) * B(4x16,f32) + C(16x16,f32)
  return __builtin_amdgcn_wmma_f32_16x16x4_f32(
      /*neg_a=*/false, a, /*neg_b=*/false, b,
      /*c_mod=*/(short)0, c, /*reuse_a=*/false, /*reuse_b=*/false);
}

// ---------------------------------------------------------------------------
// Precompute per-neuron constants: mumu[n] = ||mu_n||^2, mv[n,k] = mu_n . v_nk
// One wave per n.
// ---------------------------------------------------------------------------
__global__ __launch_bounds__(256) void precomp_w_kernel(
    const float* __restrict__ Mu,   // (N, D)
    const float* __restrict__ V,    // (N, K, D)
    float* __restrict__ mumu,       // (N)
    float* __restrict__ mv)         // (N, K)
{
  int wave = blockIdx.x * (blockDim.x >> 5) + (threadIdx.x >> 5);
  int lane = threadIdx.x & 31;
  if (wave >= NB) return;
  const float* mu = Mu + (size_t)wave * DIM;
  const float* vv = V + (size_t)wave * KK * DIM;
  float mm = 0.f;
  float pv[KK];
#pragma unroll
  for (int k = 0; k < KK; ++k) pv[k] = 0.f;
  for (int d = lane; d < DIM; d += 32) {
    float m = mu[d];
    mm += m * m;
#pragma unroll
    for (int k = 0; k < KK; ++k) pv[k] += m * vv[k * DIM + d];
  }
#pragma unroll
  for (int off = 16; off >= 1; off >>= 1) {
    mm += __shfl_xor(mm, off, 32);
#pragma unroll
    for (int k = 0; k < KK; ++k) pv[k] += __shfl_xor(pv[k], off, 32);
  }
  if (lane == 0) {
    mumu[wave] = mm;
#pragma unroll
    for (int k = 0; k < KK; ++k) mv[wave * KK + k] = pv[k];
  }
}

// ---------------------------------------------------------------------------
// xx[b] = ||x_b||^2 ; one wave per row.
// ---------------------------------------------------------------------------
__global__ __launch_bounds__(256) void rowsumsq_kernel(
    const float* __restrict__ X, float* __restrict__ xx)
{
  int wave = blockIdx.x * (blockDim.x >> 5) + (threadIdx.x >> 5);
  int lane = threadIdx.x & 31;
  if (wave >= NB) return;
  const float* xr = X + (size_t)wave * DIM;
  float s = 0.f;
  for (int d = lane; d < DIM; d += 32) { float v = xr[d]; s += v * v; }
#pragma unroll
  for (int off = 16; off >= 1; off >>= 1) s += __shfl_xor(s, off, 32);
  if (lane == 0) xx[wave] = s;
}

// ---------------------------------------------------------------------------
// Fused HMU layer:
//   H[b,n] = f( lam[n]*(xx[b] - 2 x.mu_n + mumu[n]) + sum_k (x.v_nk - mv[n,k])^2 )
// One wave per 32(b) x 16(n) tile; 18 fp32 WMMA accumulators:
//   accS[h]          : S tiles = X @ Mu^T (row halves h=0,1)
//   accP[h][0..7]    : 32x128 strip of P = X @ Vflat^T  (j = 8n+k)
// ---------------------------------------------------------------------------
__global__ __launch_bounds__(32) void hmu_gemm_kernel(
    const float* __restrict__ X,     // (B, D)
    const float* __restrict__ Mu,    // (N, D)
    const float* __restrict__ V,     // (N*K, D) row-major (natural layout of vs[i])
    const float* __restrict__ lam,   // (N)
    const float* __restrict__ xx,    // (B)
    const float* __restrict__ mumu,  // (N)
    const float* __restrict__ mv,    // (N, K)
    float* __restrict__ H,           // (B, N)
    int use_exp)
{
  __shared__ float pbuf[32 * 128];   // 32 local rows x 128 local j columns (16 KB)

  const int b0    = blockIdx.x * 32;
  const int n0    = blockIdx.y * 16;
  const int lane  = threadIdx.x;
  const int l16   = lane & 15;
  const int lhalf = lane >> 4;       // 0: K%4 in {0,1}, 1: K%4 in {2,3}

  const float* xrow  = X  + (size_t)(b0 + l16) * DIM + 2 * lhalf;
  const float* murow = Mu + (size_t)(n0 + l16) * DIM + 2 * lhalf;
  const float* vrow  = V  + (size_t)(n0 * KK + l16) * DIM + 2 * lhalf;

  const v8f vzero = {0.f, 0.f, 0.f, 0.f, 0.f, 0.f, 0.f, 0.f};
  v8f accS[2];
  v8f accP[2][8];
#pragma unroll
  for (int h = 0; h < 2; ++h) {
    accS[h] = vzero;
#pragma unroll
    for (int t = 0; t < 8; ++t) accP[h][t] = vzero;
  }

  for (int k0 = 0; k0 < DIM; k0 += 4) {
    v2f a0 = *(const v2f*)(xrow + k0);                   // rows b0 .. b0+15
    v2f a1 = *(const v2f*)(xrow + (size_t)16 * DIM + k0);// rows b0+16 .. b0+31
    v2f bm = *(const v2f*)(murow + k0);
    accS[0] = wmma_f32_4(a0, bm, accS[0]);
    accS[1] = wmma_f32_4(a1, bm, accS[1]);
#pragma unroll
    for (int t = 0; t < 8; ++t) {
      v2f bv = *(const v2f*)(vrow + (size_t)t * 16 * DIM + k0);
      accP[0][t] = wmma_f32_4(a0, bv, accP[0][t]);
      accP[1][t] = wmma_f32_4(a1, bv, accP[1][t]);
    }
  }

  // Stage P strips to LDS: row = local b, col = local j (= 8*(n-n0)+k)
#pragma unroll
  for (int h = 0; h < 2; ++h) {
#pragma unroll
    for (int t = 0; t < 8; ++t) {
#pragma unroll
      for (int r = 0; r < 8; ++r) {
        pbuf[(16 * h + r + 8 * lhalf) * 128 + t * 16 + l16] = accP[h][t][r];
      }
    }
  }
  __syncthreads();

  const int   n     = n0 + l16;
  const float lam_n = lam[n];
  const float mm_n  = mumu[n];
  float mvk[KK];
#pragma unroll
  for (int k = 0; k < KK; ++k) mvk[k] = mv[n * KK + k];

#pragma unroll
  for (int h = 0; h < 2; ++h) {
#pragma unroll
    for (int r = 0; r < 8; ++r) {
      const int   brow = b0 + 16 * h + r + 8 * lhalf;   // accumulator row mapping
      const float s    = accS[h][r];                    // x_b . mu_n
      const float sq   = xx[brow] - 2.f * s + mm_n;
      float sumk = 0.f;
      const float* prow = &pbuf[(16 * h + r + 8 * lhalf) * 128 + 8 * l16];
#pragma unroll
      for (int k = 0; k < KK; ++k) {
        float d = prow[k] - mvk[k];
        sumk += d * d;
      }
      float quad = lam_n * sq + sumk;
      float hval = use_exp ? __expf(-quad * (1.0f / (float)DIM)) : -quad;
      H[(size_t)brow * NB + n] = hval;
    }
  }
}

// ---------------------------------------------------------------------------
// BatchNorm (training mode, biased stats): per-column mean/var over B=1024.
// scale[n] = gamma/sqrt(var+eps), shift[n] = beta - mean*scale
// ---------------------------------------------------------------------------
__global__ __launch_bounds__(256) void bn_stats_kernel(
    const float* __restrict__ H,
    const float* __restrict__ gamma,
    const float* __restrict__ beta,
    float* __restrict__ scale,
    float* __restrict__ shift)
{
  int n = blockIdx.x * blockDim.x + threadIdx.x;
  if (n >= NB) return;
  float s = 0.f, s2 = 0.f;
  for (int b = 0; b < 1024; ++b) {
    float h = H[(size_t)b * NB + n];
    s += h;
    s2 += h * h;
  }
  float mean = s * (1.0f / 1024.0f);
  float var  = s2 * (1.0f / 1024.0f) - mean * mean;
  float inv  = rsqrtf(var + BN_EPS);
  float sc   = gamma[n] * inv;
  scale[n] = sc;
  shift[n] = beta[n] - mean * sc;
}

__global__ __launch_bounds__(256) void bn_apply_kernel(
    const float* __restrict__ H,
    const float* __restrict__ scale,
    const float* __restrict__ shift,
    float* __restrict__ out)
{
  int i = blockIdx.x * blockDim.x + threadIdx.x;   // 1M elements
  int n = i & (NB - 1);
  out[i] = H[i] * scale[n] + shift[n];
}

// ---------------------------------------------------------------------------
// logits[b,c] = relu(F[b,:]) @ Wc[c,:] + bc[c].  One block per row b.
// ---------------------------------------------------------------------------
__global__ __launch_bounds__(256) void logits_kernel(
    const float* __restrict__ F,     // (B, N)
    const float* __restrict__ Wc,    // (C, N)
    const float* __restrict__ bc,    // (C)
    float* __restrict__ out)         // (B, C)
{
  __shared__ float red[256];
  const int b = blockIdx.x;
  const int tid = threadIdx.x;
  float acc[NCLS];
#pragma unroll
  for (int c = 0; c < NCLS; ++c) acc[c] = 0.f;
  for (int n = tid; n < NB; n += 256) {
    float xv = F[(size_t)b * NB + n];
    xv = xv > 0.f ? xv : 0.f;
#pragma unroll
    for (int c = 0; c < NCLS; ++c) acc[c] += xv * Wc[c * NB + n];
  }
#pragma unroll
  for (int c = 0; c < NCLS; ++c) {
    red[tid] = acc[c];
    __syncthreads();
    for (int s = 128; s > 0; s >>= 1) {
      if (tid < s) red[tid] += red[tid + s];
      __syncthreads();
    }
    if (tid == 0) out[b * NCLS + c] = red[0] + bc[c];
    __syncthreads();
  }
}

// ---------------------------------------------------------------------------
// Host launcher
// ---------------------------------------------------------------------------
extern "C" void kernel_launch(void* const* d_in, const int* in_sizes, int n_in,
                              void* d_out, int out_size, void* d_ws, size_t ws_size,
                              hipStream_t stream) {
  (void)in_sizes; (void)n_in; (void)out_size; (void)ws_size;

  const float* x    = (const float*)d_in[0];   // (1024,1024)
  const float* mus  = (const float*)d_in[1];   // (3,1024,1024)
  const float* lams = (const float*)d_in[2];   // (3,1024)
  const float* vs   = (const float*)d_in[3];   // (3,1024,8,1024)
  const float* gam  = (const float*)d_in[4];   // (3,1024)
  const float* bet  = (const float*)d_in[5];   // (3,1024)
  const float* Wc   = (const float*)d_in[6];   // (10,1024)
  const float* bc   = (const float*)d_in[7];   // (10)

  float* out    = (float*)d_out;
  float* logits = out;                                  // 1024*10
  float* feats0 = out + 1024 * NCLS;                    // 1024*1024 each
  float* feats1 = feats0 + 1024 * 1024;
  float* feats2 = feats1 + 1024 * 1024;
  float* feats[3] = {feats0, feats1, feats2};

  float* H     = (float*)d_ws;                          // 1024*1024
  float* xx    = H + 1024 * 1024;                       // 1024
  float* mumu  = xx + 1024;                             // 1024
  float* mv    = mumu + 1024;                           // 1024*8
  float* scale = mv + 1024 * KK;                        // 1024
  float* shift = scale + 1024;                          // 1024

  const float* cur = x;
  for (int i = 0; i < 3; ++i) {
    const float* Mu = mus + (size_t)i * 1024 * 1024;
    const float* V  = vs  + (size_t)i * 1024 * KK * 1024;
    precomp_w_kernel<<<128, 256, 0, stream>>>(Mu, V, mumu, mv);
    rowsumsq_kernel<<<128, 256, 0, stream>>>(cur, xx);
    hmu_gemm_kernel<<<dim3(32, 64), 32, 0, stream>>>(
        cur, Mu, V, lams + i * 1024, xx, mumu, mv, H, (i == 2) ? 1 : 0);
    bn_stats_kernel<<<4, 256, 0, stream>>>(H, gam + i * 1024, bet + i * 1024,
                                           scale, shift);
    bn_apply_kernel<<<4096, 256, 0, stream>>>(H, scale, shift, feats[i]);
    cur = feats[i];
  }
  logits_kernel<<<1024, 256, 0, stream>>>(feats[2], Wc, bc, logits);
}